// GraphSage_9302899163518
// MI455X (gfx1250) — compile-verified
//
#include <hip/hip_runtime.h>

typedef __attribute__((ext_vector_type(2))) float v2f;
typedef __attribute__((ext_vector_type(4))) float v4f;
typedef __attribute__((ext_vector_type(8))) float v8f;

#define DIN        256
#define DCOMB      512
#define DOUT       256
#define KNEIGH     10
#define ROWS_PER_WG 64
#define LDS_STRIDE 516   // 512 + 4 dword pad -> conflict-free ds_load_b64 A-frags
#define THREADS    256

extern "C" __global__ __launch_bounds__(THREADS)
void sage_layer_kernel(const float* __restrict__ embs,     // [n_src, 256]
                       const int*   __restrict__ self_idx, // [n]
                       const int*   __restrict__ neigh_idx,// [n, 10]
                       const float* __restrict__ W,        // [256, 512] row-major
                       float*       __restrict__ out,      // [n, 256]
                       int n)
{
    extern __shared__ float lds[];  // ROWS_PER_WG * LDS_STRIDE floats
    const int tid     = threadIdx.x;
    const int wg_row0 = blockIdx.x * ROWS_PER_WG;

    // ---------------- Phase 1: gather + mean into LDS combined tile ----------------
    // 256 threads = 16 rows x 16 col-threads; 4 passes over 64 rows.
    const int rlocal = tid >> 4;   // 0..15
    const int c16    = tid & 15;   // 0..15, each owns 16 floats as 4 strided v4f
    for (int pass = 0; pass < 4; ++pass) {
        const int row  = pass * 16 + rlocal;      // local row 0..63
        const int grow = wg_row0 + row;
        float* dst = &lds[row * LDS_STRIDE];
        if (grow < n) {
            // self half: cols [0,256)
            const float* srow = embs + (size_t)self_idx[grow] * DIN;
            #pragma unroll
            for (int j = 0; j < 4; ++j) {
                const int col = c16 * 4 + j * 64;   // lanes 0..15 cover 256B/instr
                *(v4f*)(dst + col) = *(const v4f*)(srow + col);
            }
            // aggregate half: cols [256,512) = mean over 10 neighbors
            v4f acc[4];
            #pragma unroll
            for (int j = 0; j < 4; ++j) acc[j] = (v4f){0.f, 0.f, 0.f, 0.f};
            const int* nrow = neigh_idx + (size_t)grow * KNEIGH;
            for (int kk = 0; kk < KNEIGH; ++kk) {
                const float* frow = embs + (size_t)nrow[kk] * DIN;
                #pragma unroll
                for (int j = 0; j < 4; ++j) {
                    const int col = c16 * 4 + j * 64;
                    acc[j] += *(const v4f*)(frow + col);
                }
            }
            const float inv = 1.0f / (float)KNEIGH;
            #pragma unroll
            for (int j = 0; j < 4; ++j) {
                const int col = c16 * 4 + j * 64;
                *(v4f*)(dst + DIN + col) = acc[j] * inv;
            }
        } else {
            // zero-fill tail rows so WMMA reads are defined
            #pragma unroll
            for (int j = 0; j < 4; ++j) {
                const int col = c16 * 4 + j * 64;
                *(v4f*)(dst + col)       = (v4f){0.f, 0.f, 0.f, 0.f};
                *(v4f*)(dst + DIN + col) = (v4f){0.f, 0.f, 0.f, 0.f};
            }
        }
    }
    __syncthreads();

    // ---------------- Phase 2: WMMA GEMM  out = relu(combined @ W^T) ----------------
    // 8 waves; wave w owns N-tiles {2w, 2w+1}; 4 M-tiles of 16 rows each.
    const int wave  = tid >> 5;
    const int lane  = tid & 31;
    const int lhalf = lane >> 4;   // 0: K=k0,k0+1  1: K=k0+2,k0+3 (A/B frag halves)
    const int l16   = lane & 15;

    for (int t = 0; t < 2; ++t) {
        const int nt = wave * 2 + t;                       // N-tile index 0..15
        // B frag source: B[k][o] = W[o][k]; this lane serves column o = nt*16+l16
        const float* wrow = W + (size_t)(nt * 16 + l16) * DCOMB + 2 * lhalf;

        v8f acc[4];
        #pragma unroll
        for (int m = 0; m < 4; ++m)
            acc[m] = (v8f){0.f, 0.f, 0.f, 0.f, 0.f, 0.f, 0.f, 0.f};

        const float* abase = &lds[l16 * LDS_STRIDE + 2 * lhalf];

        for (int k0 = 0; k0 < DCOMB; k0 += 4) {
            const v2f b = *(const v2f*)(wrow + k0);        // global b64, reused 4x
            #pragma unroll
            for (int m = 0; m < 4; ++m) {
                const v2f a = *(const v2f*)(abase + m * 16 * LDS_STRIDE + k0); // ds b64
                acc[m] = __builtin_amdgcn_wmma_f32_16x16x4_f32(
                    /*neg_a=*/false, a, /*neg_b=*/false, b,
                    /*c_mod=*/(short)0, acc[m],
                    /*reuse_a=*/false, /*reuse_b=*/false);
            }
        }

        // C/D layout: VGPR r -> M = r + 8*lhalf, N = l16 (within tile)
        #pragma unroll
        for (int m = 0; m < 4; ++m) {
            const int grow_base = wg_row0 + m * 16 + 8 * lhalf;
            #pragma unroll
            for (int r = 0; r < 8; ++r) {
                const int grow = grow_base + r;
                if (grow < n)
                    out[(size_t)grow * DOUT + nt * 16 + l16] = fmaxf(acc[m][r], 0.f);
            }
        }
    }
}

extern "C" void kernel_launch(void* const* d_in, const int* in_sizes, int n_in,
                              void* d_out, int out_size, void* d_ws, size_t ws_size,
                              hipStream_t stream) {
    const float* raw    = (const float*)d_in[0];
    const float* W1     = (const float*)d_in[1];
    const float* W2     = (const float*)d_in[2];
    const int*   nodes1 = (const int*)  d_in[3];
    const int*   neigh1 = (const int*)  d_in[4];
    const int*   self2  = (const int*)  d_in[5];
    const int*   neigh2 = (const int*)  d_in[6];

    const int N1 = in_sizes[3];   // 100000
    const int N2 = in_sizes[5];   // 20000

    float* h1  = (float*)d_ws;    // [N1, 256] intermediate
    float* out = (float*)d_out;   // [N2, 256]

    const size_t lds_bytes = (size_t)ROWS_PER_WG * LDS_STRIDE * sizeof(float); // ~129 KB

    dim3 block(THREADS);
    dim3 grid1((N1 + ROWS_PER_WG - 1) / ROWS_PER_WG);
    dim3 grid2((N2 + ROWS_PER_WG - 1) / ROWS_PER_WG);

    sage_layer_kernel<<<grid1, block, lds_bytes, stream>>>(raw, nodes1, neigh1, W1, h1, N1);
    sage_layer_kernel<<<grid2, block, lds_bytes, stream>>>(h1,  self2,  neigh2, W2, out, N2);
}